// RFAConv_66503273611866
// MI455X (gfx1250) — compile-verified
//
#include <hip/hip_runtime.h>
#include <hip/hip_bf16.h>

typedef float v2f __attribute__((ext_vector_type(2)));
typedef float v8f __attribute__((ext_vector_type(8)));

#define EPSV 1e-5f
#define Bn 8
#define Cn 64
#define Hn 128
#define Wn 128
#define On 128
#define KK 9
#define KDIM 576          // Cn * KK
#define TILE 32           // pixels per workgroup (32 consecutive w)
#define ROWSTRIDE 580     // prodA row stride: 580 % 64 == 4 -> conflict-free b64 A reads

// workspace layout (float offsets)
#define WS_W1F   0        // [Cn][9][9]  BN1-fused grouped weights     (5184)
#define WS_B1F   5184     // [Cn][9]     BN1-fused bias                 (576)
#define WS_CBF   5760     // [On]        BN2-fused conv bias            (128)
#define WS_BPACK 5888     // [144][8][32][2] WMMA-B-fragment conv_w   (73728)
#define WS_TOTAL 79616

__global__ __launch_bounds__(256) void rfa_prep(
    const float* __restrict__ w1, const float* __restrict__ b1,
    const float* __restrict__ g1, const float* __restrict__ beta1,
    const float* __restrict__ m1, const float* __restrict__ v1,
    const float* __restrict__ conv_w, const float* __restrict__ conv_b,
    const float* __restrict__ g2, const float* __restrict__ beta2,
    const float* __restrict__ m2, const float* __restrict__ v2,
    float* __restrict__ ws)
{
    const int i = blockIdx.x * blockDim.x + threadIdx.x;
    if (i < 5184) {                       // w1f[c][o][k] = w1 * g1/sqrt(v1+eps)
        const int c = i / 81, o = (i % 81) / 9;
        const float s = g1[c * 9 + o] * rsqrtf(v1[c * 9 + o] + EPSV);
        ws[WS_W1F + i] = w1[i] * s;
    }
    const int j = i - 5184;
    if (j >= 0 && j < 576) {              // b1f = (b1 - m1)*s1 + beta1
        const float s = g1[j] * rsqrtf(v1[j] + EPSV);
        ws[WS_B1F + j] = (b1[j] - m1[j]) * s + beta1[j];
    }
    const int q = i - 5760;
    if (q >= 0 && q < 128) {              // cbf = (conv_b - m2)*s2 + beta2
        const float s = g2[q] * rsqrtf(v2[q] + EPSV);
        ws[WS_CBF + q] = (conv_b[q] - m2[q]) * s + beta2[q];
    }
    const int r = i - 5888;
    if (r >= 0 && r < 73728) {            // bpack[(kb*8+nt)*64 + lane*2 + jj]
        const int jj   = r & 1;
        const int lane = (r >> 1) & 31;
        const int nt   = (r >> 6) & 7;
        const int kb   = r >> 9;
        const int krow = kb * 4 + ((lane >> 4) << 1) + jj;   // K index
        const int n    = nt * 16 + (lane & 15);              // output channel
        const float s  = g2[n] * rsqrtf(v2[n] + EPSV);
        ws[WS_BPACK + r] = conv_w[n * KDIM + krow] * s;
    }
}

__global__ __launch_bounds__(256) void rfa_main(
    const float* __restrict__ x, const float* __restrict__ ws,
    float* __restrict__ out)
{
    const float* __restrict__ w1f   = ws + WS_W1F;
    const float* __restrict__ b1f   = ws + WS_B1F;
    const float* __restrict__ cbf   = ws + WS_CBF;
    const float* __restrict__ bpack = ws + WS_BPACK;

    __shared__ float w1s[Cn * 81];
    __shared__ float b1s[Cn * KK];
    __shared__ float prodA[TILE * ROWSTRIDE];

    const int tid  = threadIdx.x;
    const int tile = blockIdx.x;                 // Bn*Hn*Wn/32 tiles
    const int b    = tile >> 9;                  // 512 tiles per image
    const int h    = (tile & 511) >> 2;          // 4 tiles per row
    const int w0   = (tile & 3) << 5;            // *32

    for (int i = tid; i < Cn * 81; i += 256) w1s[i] = w1f[i];
    for (int i = tid; i < Cn * KK; i += 256) b1s[i] = b1f[i];
    __syncthreads();

    // ---------------- Stage A: per-channel 9x9 + softmax + reweight ----------
    {
        const int p  = tid & 31;                 // pixel within tile (coalesced w)
        const int c0 = tid >> 5;                 // base channel (uniform per wave)
        const int wg = w0 + p;
        #pragma unroll
        for (int t = 0; t < 8; ++t) {
            const int c = c0 + 8 * t;
            const float* __restrict__ xc = x + ((size_t)(b * Cn + c) * Hn) * Wn;
            float patch[KK];
            #pragma unroll
            for (int di = 0; di < 3; ++di) {
                const int hh = h + di - 1;
                #pragma unroll
                for (int dj = 0; dj < 3; ++dj) {
                    const int wwp = wg + dj - 1;
                    float v = 0.0f;
                    if (hh >= 0 && hh < Hn && wwp >= 0 && wwp < Wn)
                        v = xc[hh * Wn + wwp];
                    patch[di * 3 + dj] = v;
                }
            }
            const float* __restrict__ wr = &w1s[c * 81];  // full-wave broadcast
            const float* __restrict__ br = &b1s[c * KK];
            float wt[KK];
            #pragma unroll
            for (int o = 0; o < KK; ++o) {
                float acc = br[o];
                #pragma unroll
                for (int k = 0; k < KK; ++k)
                    acc = fmaf(wr[o * KK + k], patch[k], acc);
                wt[o] = acc;
            }
            float mx = wt[0];
            #pragma unroll
            for (int o = 1; o < KK; ++o) mx = fmaxf(mx, wt[o]);
            float s = 0.0f;
            #pragma unroll
            for (int o = 0; o < KK; ++o) { wt[o] = __expf(wt[o] - mx); s += wt[o]; }
            const float inv = 1.0f / s;
            float* __restrict__ dst = &prodA[p * ROWSTRIDE + c * KK];
            #pragma unroll
            for (int o = 0; o < KK; ++o) dst[o] = patch[o] * wt[o] * inv;
        }
    }
    __syncthreads();

    // -------- Stage B: [32x576] x [576x16] per wave, dual-M f32 WMMA ---------
    {
        const int lane  = tid & 31;
        const int wave  = tid >> 5;              // 8 waves -> 8 N-tiles of 16
        const int m     = lane & 15;
        const int khalf = (lane >> 4) << 1;      // 0 | 2  (K split per lane-half)
        const float* __restrict__ arow0 = &prodA[m * ROWSTRIDE];
        const float* __restrict__ arow1 = &prodA[(m + 16) * ROWSTRIDE];
        const float* __restrict__ bbase = bpack + wave * 64 + lane * 2;

        v8f acc0 = {};
        v8f acc1 = {};
        #pragma unroll 4
        for (int kb = 0; kb < KDIM / 4; ++kb) {
            const int k0 = kb * 4 + khalf;
            v2f bf = *(const v2f*)(bbase + kb * 512);   // coalesced 256B/wave
            v2f a0 = *(const v2f*)(arow0 + k0);         // ds_load_b64, bank-clean
            v2f a1 = *(const v2f*)(arow1 + k0);
            acc0 = __builtin_amdgcn_wmma_f32_16x16x4_f32(
                false, a0, false, bf, (short)0, acc0, false, false);
            acc1 = __builtin_amdgcn_wmma_f32_16x16x4_f32(
                false, a1, false, bf, (short)0, acc1, false, false);
        }

        const int o    = wave * 16 + m;          // output channel (N)
        const float bs = cbf[o];
        const int mb   = (lane >> 4) << 3;       // D rows: 0..7 | 8..15
        float* __restrict__ op0 =
            out + (((size_t)(b * On + o) * Hn + h) * Wn) + w0 + mb;
        #pragma unroll
        for (int r = 0; r < 8; ++r) {
            const float v = acc0[r] + bs;
            op0[r] = v > 0.0f ? v : 0.0f;
        }
        float* __restrict__ op1 = op0 + 16;      // pixels w0+16 .. w0+31
        #pragma unroll
        for (int r = 0; r < 8; ++r) {
            const float v = acc1[r] + bs;
            op1[r] = v > 0.0f ? v : 0.0f;
        }
    }
}

extern "C" void kernel_launch(void* const* d_in, const int* in_sizes, int n_in,
                              void* d_out, int out_size, void* d_ws, size_t ws_size,
                              hipStream_t stream)
{
    const float* x      = (const float*)d_in[0];
    const float* w1     = (const float*)d_in[1];
    const float* b1     = (const float*)d_in[2];
    const float* g1     = (const float*)d_in[3];
    const float* beta1  = (const float*)d_in[4];
    const float* m1     = (const float*)d_in[5];
    const float* v1     = (const float*)d_in[6];
    const float* conv_w = (const float*)d_in[7];
    const float* conv_b = (const float*)d_in[8];
    const float* g2     = (const float*)d_in[9];
    const float* beta2  = (const float*)d_in[10];
    const float* m2     = (const float*)d_in[11];
    const float* v2     = (const float*)d_in[12];

    float* ws  = (float*)d_ws;
    float* out = (float*)d_out;

    rfa_prep<<<(WS_TOTAL + 255) / 256, 256, 0, stream>>>(
        w1, b1, g1, beta1, m1, v1, conv_w, conv_b, g2, beta2, m2, v2, ws);

    const int ntiles = Bn * Hn * Wn / TILE;      // 4096
    rfa_main<<<ntiles, 256, 0, stream>>>(x, ws, out);
}